// Detect_44633300140475
// MI455X (gfx1250) — compile-verified
//
#include <hip/hip_runtime.h>
#include <math.h>

// Problem constants (match reference)
#define BB 32
#define PP 8732
#define CC 81
#define KK 200
#define KP 208          // K padded to 13*16 for WMMA tiling
#define NWORD 7         // 224 bits >= 208 candidate columns
#define PADP 8960       // PP padded to 35*256
#define CHUNK 35        // PADP / 256
#define NTHREADS 256
#define CONF_T 0.01f
#define NMS_T 0.45f

typedef __attribute__((ext_vector_type(2))) float v2f;
typedef __attribute__((ext_vector_type(8))) float v8f;

// ---------------------------------------------------------------------------
// CDNA5 async DMA: global -> LDS, 4B per lane, tracked by ASYNCcnt.
// GVS mode: mem = SGPR64 base + VGPR32 byte offset. VDST VGPR holds the LDS
// byte offset (low 32 bits of a generic __shared__ pointer per the aperture
// mapping).
// ---------------------------------------------------------------------------
__device__ __forceinline__ void async_load_f32_to_lds(const float* gbase,
                                                      unsigned byte_off,
                                                      const float* lds_dst) {
  unsigned lds_off = (unsigned)(uintptr_t)lds_dst;
  asm volatile("global_load_async_to_lds_b32 %0, %1, %2"
               :
               : "v"(lds_off), "v"(byte_off), "s"(gbase)
               : "memory");
}
__device__ __forceinline__ void wait_asynccnt0() {
  asm volatile("s_wait_asynccnt 0" ::: "memory");
}

// ---------------------------------------------------------------------------
// Tiled transpose: conf[B*P, C] -> confT[B, C, P] so NMS WGs read contiguous
// score rows instead of stride-324B columns (kills 32x cacheline blowup).
// ---------------------------------------------------------------------------
__global__ __launch_bounds__(256) void detect_transpose_kernel(
    const float* __restrict__ conf, float* __restrict__ confT)
{
  __shared__ float tile[32][33];
  const int bp0 = blockIdx.x * 32;
  const int c0  = blockIdx.y * 32;
  const int tx = threadIdx.x, ty = threadIdx.y;   // 32 x 8
#pragma unroll
  for (int i = 0; i < 32; i += 8) {
    int bp = bp0 + ty + i, cc = c0 + tx;
    float v = 0.0f;
    if (bp < BB * PP && cc < CC) v = conf[(size_t)bp * CC + cc];
    tile[ty + i][tx] = v;
  }
  __syncthreads();
#pragma unroll
  for (int i = 0; i < 32; i += 8) {
    int cc = c0 + ty + i, bp = bp0 + tx;
    if (cc < CC && bp < BB * PP) {
      int b = bp / PP, p = bp - b * PP;
      confT[((size_t)b * CC + cc) * PP + p] = tile[tx][ty + i];
    }
  }
}

// ---------------------------------------------------------------------------
// One workgroup (256 threads = 8 wave32) per (batch, class) pair.
// ---------------------------------------------------------------------------
__global__ __launch_bounds__(NTHREADS) void detect_nms_kernel(
    const float* __restrict__ loc, const float* __restrict__ conf,
    const float* __restrict__ priors, const float* __restrict__ confT,
    int use_t, float* __restrict__ out)
{
  const int blk = blockIdx.x;
  const int b = blk / CC;
  const int c = blk % CC;
  const int tid = threadIdx.x;
  float* outBlk = out + (size_t)blk * (KK * 5);

  // class 0 == background: all zeros
  if (c == 0) {
    for (int i = tid; i < KK * 5; i += NTHREADS) outBlk[i] = 0.0f;
    return;
  }

  __shared__ float    sc[PADP];          // scores (thresholded in-place)
  __shared__ float    topv[KP];          // top-K scores (sorted desc)
  __shared__ int      topi[KP];          // top-K prior indices
  __shared__ float4   bxs[KP];           // decoded corner boxes
  __shared__ float    areas[KP];
  __shared__ unsigned smask[KP * NWORD]; // suppression bitmask rows
  __shared__ float    wrv[8];            // per-wave reduction values
  __shared__ int      wri[8];
  __shared__ unsigned keepw[NWORD];

  // ---- 1) DMA scores straight into LDS (ASYNCcnt-tracked) -----------------
  {
    const float*   gb  = use_t ? (confT + ((size_t)b * CC + c) * PP)
                               : (conf + (size_t)b * PP * CC + c);
    const unsigned esz = use_t ? 4u : (unsigned)(CC * 4);
#pragma unroll 1
    for (int k = 0; k < CHUNK; ++k) {
      int p = tid + (k << 8);
      if (p < PP) async_load_f32_to_lds(gb, (unsigned)p * esz, &sc[p]);
      else        sc[p] = 0.0f;          // pad tail
    }
    wait_asynccnt0();                    // own chunk is all this thread reads
  }
  for (int i = tid; i < KP * NWORD; i += NTHREADS) smask[i] = 0u;
  if (tid < NWORD) keepw[tid] = 0u;
  if (tid < KP - KK) { topv[KK + tid] = 0.0f; topi[KK + tid] = 0; }
  __syncthreads();

  // ---- 2) top-K tournament selection --------------------------------------
  // Each thread owns p = tid + 256k (k<35); threshold fused into first scan.
  // Strict '>' over ascending scan keeps the smallest index on ties,
  // matching jax.lax.top_k stability.
  float pv = -1.0f; int pi = tid;
  for (int k = 0; k < CHUNK; ++k) {
    int p = tid + (k << 8);
    float v = sc[p];
    v = (v > CONF_T) ? v : 0.0f;
    sc[p] = v;
    if (v > pv) { pv = v; pi = p; }
  }
  const int lane = tid & 31, wid = tid >> 5;
  for (int r = 0; r < KK; ++r) {
    float rv = pv; int ri = pi;
#pragma unroll
    for (int off = 16; off > 0; off >>= 1) {      // wave32 butterfly reduce
      float ov = __shfl_xor(rv, off, 32);
      int   oi = __shfl_xor(ri, off, 32);
      if (ov > rv || (ov == rv && oi < ri)) { rv = ov; ri = oi; }
    }
    if (lane == 0) { wrv[wid] = rv; wri[wid] = ri; }
    __syncthreads();
    float gv = wrv[0]; int gi = wri[0];
#pragma unroll
    for (int w = 1; w < 8; ++w) {
      float ov = wrv[w]; int oi = wri[w];
      if (ov > gv || (ov == gv && oi < gi)) { gv = ov; gi = oi; }
    }
    if (pi == gi) {                               // unique owner thread
      topv[r] = gv; topi[r] = gi;
      sc[gi] = -1.0f;
      pv = -1.0f; pi = tid;                       // rescan own chunk only
      for (int k = 0; k < CHUNK; ++k) {
        int p = tid + (k << 8);
        float v = sc[p];
        if (v > pv) { pv = v; pi = p; }
      }
    }
    __syncthreads();
  }

  // ---- 3) decode boxes for the 200 winners (pads -> zero box) -------------
  for (int j = tid; j < KP; j += NTHREADS) {
    float4 box = make_float4(0.f, 0.f, 0.f, 0.f);
    float v = topv[j]; int i = topi[j];
    if (j < KK && v > CONF_T && i < PP) {
      const float4 l4 = ((const float4*)loc)[(size_t)b * PP + i];
      const float4 p4 = ((const float4*)priors)[i];
      float cx = p4.x + l4.x * 0.1f * p4.z;
      float cy = p4.y + l4.y * 0.1f * p4.w;
      float w  = p4.z * expf(l4.z * 0.2f);
      float h  = p4.w * expf(l4.w * 0.2f);
      box.x = cx - 0.5f * w; box.y = cy - 0.5f * h;
      box.z = box.x + w;     box.w = box.y + h;
      atomicOr(&keepw[j >> 5], 1u << (j & 31));   // keep init: score > thresh
    }
    bxs[j] = box;
    areas[j] = (box.z - box.x) * (box.w - box.y);
  }
  __syncthreads();

  // ---- 4) pairwise suppression bitmask; WMMA computes t*(area_i+area_j) ---
  // D[m][n] = (t*area[m])*1 + 1*(t*area[n]) via V_WMMA_F32_16X16X4_F32.
  // iou > t  <=>  inter*(1+t) > t*(ai+aj)  (union >= 0; pad 0>0 -> false),
  // so the epilogue is division-free: one mul + compare per pair.
  for (int tile = wid; tile < 13 * 13; tile += NTHREADS / 32) {  // wave-uniform
    int ti = tile / 13, tj = tile % 13;
    float lo16 = (lane < 16) ? 1.0f : 0.0f;
    v2f A;  A.x  = lo16 * (NMS_T * areas[ti * 16 + (lane & 15)]); A.y = lo16;
    v2f Bv; Bv.x = lo16; Bv.y = lo16 * (NMS_T * areas[tj * 16 + (lane & 15)]);
    v8f Cz = {0.f, 0.f, 0.f, 0.f, 0.f, 0.f, 0.f, 0.f};
    v8f D = __builtin_amdgcn_wmma_f32_16x16x4_f32(
        /*neg_a=*/false, A, /*neg_b=*/false, Bv,
        /*c_mod=*/(short)0, Cz, /*reuse_a=*/false, /*reuse_b=*/false);
    int jn = tj * 16 + (lane & 15);               // D col N = lane&15
    int ibase = ti * 16 + ((lane >> 4) << 3);     // D row M = e + 8*(lane>=16)
    float4 bj = bxs[jn];
#pragma unroll
    for (int e = 0; e < 8; ++e) {
      int im = ibase + e;
      float4 bi = bxs[im];
      float iw = fminf(bi.z, bj.z) - fmaxf(bi.x, bj.x);
      float ih = fminf(bi.w, bj.w) - fmaxf(bi.y, bj.y);
      float inter = fmaxf(iw, 0.0f) * fmaxf(ih, 0.0f);
      if (inter * (1.0f + NMS_T) > D[e])
        atomicOr(&smask[im * NWORD + (jn >> 5)], 1u << (jn & 31));
    }
  }
  __syncthreads();

  // ---- 5) greedy serial NMS (thread 0, ~2k scalar ops) --------------------
  if (tid == 0) {
#pragma unroll 1
    for (int i = 0; i < KK; ++i) {
      int wi = i >> 5, bi = i & 31;
      if (!((keepw[wi] >> bi) & 1u)) continue;
      unsigned hiMask = (bi == 31) ? 0u : (0xFFFFFFFFu << (bi + 1)); // only j>i
#pragma unroll
      for (int w = 0; w < NWORD; ++w) {
        unsigned gate = (w < wi) ? 0u : ((w == wi) ? hiMask : 0xFFFFFFFFu);
        keepw[w] &= ~(smask[i * NWORD + w] & gate);
      }
    }
  }
  __syncthreads();

  // ---- 6) rank-by-popcount compaction + zero fill -------------------------
  int nk = 0;
#pragma unroll
  for (int w = 0; w < NWORD; ++w) nk += __popc(keepw[w]);
  for (int j = tid; j < KK; j += NTHREADS) {
    if ((keepw[j >> 5] >> (j & 31)) & 1u) {
      int rank = 0;
#pragma unroll
      for (int w = 0; w < NWORD; ++w) {
        if (w < (j >> 5)) rank += __popc(keepw[w]);
        else if (w == (j >> 5)) rank += __popc(keepw[w] & ((1u << (j & 31)) - 1u));
      }
      float* row = outBlk + rank * 5;
      float4 bx4 = bxs[j];
      row[0] = topv[j]; row[1] = bx4.x; row[2] = bx4.y; row[3] = bx4.z; row[4] = bx4.w;
    }
  }
  for (int i = nk * 5 + tid; i < KK * 5; i += NTHREADS) outBlk[i] = 0.0f;
}

// ---------------------------------------------------------------------------
extern "C" void kernel_launch(void* const* d_in, const int* in_sizes, int n_in,
                              void* d_out, int out_size, void* d_ws, size_t ws_size,
                              hipStream_t stream) {
  const float* loc    = (const float*)d_in[0];   // [B,P,4]
  const float* conf   = (const float*)d_in[1];   // [B*P,C]
  const float* priors = (const float*)d_in[2];   // [P,4]
  float* out = (float*)d_out;                    // [B,C,K,5]

  const size_t needT = (size_t)BB * CC * PP * sizeof(float);  // 90.5 MB
  const int use_t = (ws_size >= needT) ? 1 : 0;  // host constant -> deterministic
  float* confT = (float*)d_ws;

  if (use_t) {
    dim3 tb(32, 8);
    dim3 tg((BB * PP + 31) / 32, (CC + 31) / 32);
    detect_transpose_kernel<<<tg, tb, 0, stream>>>(conf, confT);
  }
  detect_nms_kernel<<<BB * CC, NTHREADS, 0, stream>>>(loc, conf, priors,
                                                      confT, use_t, out);
}